// kNNRepulsionLoss_32177894981700
// MI455X (gfx1250) — compile-verified
//
#include <hip/hip_runtime.h>
#include <hip/hip_bf16.h>

typedef __attribute__((ext_vector_type(2))) float v2f;
typedef __attribute__((ext_vector_type(8))) float v8f;

#define NPTS   65536
#define NSEEDS 20
#define KNN    10            // neighbors kept
#define HBW2   (0.01f*0.01f) // h^2

// ---------------------------------------------------------------------------
// Sorted-ascending insert (branchless swap chain); arr stays in registers.
// ---------------------------------------------------------------------------
__device__ __forceinline__ void insert11(float (&arr)[11], float v) {
#pragma unroll
    for (int k = 0; k < 11; ++k) {
        float cur = arr[k];
        bool sw   = v < cur;
        arr[k]    = sw ? v : cur;
        v         = sw ? cur : v;
    }
}

// ---------------------------------------------------------------------------
// Kernel 1: farthest point sampling. One 1024-thread block per batch.
// Each thread keeps 64 min-distances in registers (strided coverage of N).
// 20 serial steps, LDS argmax tree per step. Writes seed coords to d_ws.
// ---------------------------------------------------------------------------
__global__ void __launch_bounds__(1024)
fps_kernel(const float* __restrict__ pcs, float* __restrict__ seeds,
           float* __restrict__ out) {
    const int b   = blockIdx.x;
    const int tid = threadIdx.x;
    if (b == 0 && tid == 0) out[0] = 0.0f;   // zero the loss accumulator

    const float* p = pcs + (size_t)b * NPTS * 3;

    float dm[64];
#pragma unroll
    for (int j = 0; j < 64; ++j) dm[j] = 1.0e10f;

    __shared__ float s_red[1024];
    __shared__ int   s_idx[1024];
    __shared__ float s_c[3];

    int far = 0;   // reference uses jax.random.randint(key(1)); not reproducible

    for (int s = 0; s < NSEEDS; ++s) {
        if (tid == 0) {
            float cx = p[(size_t)far * 3 + 0];
            float cy = p[(size_t)far * 3 + 1];
            float cz = p[(size_t)far * 3 + 2];
            s_c[0] = cx; s_c[1] = cy; s_c[2] = cz;
            float* sd = seeds + ((size_t)b * NSEEDS + s) * 3;
            sd[0] = cx; sd[1] = cy; sd[2] = cz;   // seeds[s] = point[far] (pre-update)
        }
        __syncthreads();
        const float cx = s_c[0], cy = s_c[1], cz = s_c[2];

        float best = -1.0e30f;
        int   bidx = 0;
#pragma unroll
        for (int j = 0; j < 64; ++j) {
            int idx = tid + j * 1024;
            const float* q = p + (size_t)idx * 3;
            float dx = q[0] - cx, dy = q[1] - cy, dz = q[2] - cz;
            float d  = dx * dx + dy * dy + dz * dz;
            float nm = fminf(dm[j], d);
            dm[j] = nm;
            if (nm > best) { best = nm; bidx = idx; }   // strict > keeps lowest idx
        }
        s_red[tid] = best;
        s_idx[tid] = bidx;
        __syncthreads();
        for (int off = 512; off > 0; off >>= 1) {
            if (tid < off) {
                float ov = s_red[tid + off];
                int   oi = s_idx[tid + off];
                if (ov > s_red[tid] || (ov == s_red[tid] && oi < s_idx[tid])) {
                    s_red[tid] = ov; s_idx[tid] = oi;
                }
            }
            __syncthreads();
        }
        far = s_idx[0];
        __syncthreads();
    }
}

// ---------------------------------------------------------------------------
// Kernel 2: fused distance (via V_WMMA_F32_16X16X4_F32) + top-11 + loss.
// grid = (2 seed-groups, B). 256 threads = 8 waves; each wave WMMAs a
// 16-point x 16-seed tile:
//   A[m]   = (-2x, -2y, -2z, |p_m|^2)   (lanes 0-15: K0,K1; lanes 16-31: K2,K3)
//   B[:,n] = ( sx,  sy,  sz, 1 )
//   D[m][n] = |p_m|^2 - 2 p.s    =>   d^2 = D + |s_n|^2  (per-lane constant)
// Top-11 selection is done on d^2 (monotone under sqrt); sqrt deferred to the
// final 10-value loss stage.
// ---------------------------------------------------------------------------
__global__ void __launch_bounds__(256)
knn_loss_kernel(const float* __restrict__ pcs, const float* __restrict__ seeds,
                float* __restrict__ out, float inv_b) {
    const int g    = blockIdx.x;         // seed group: 0 -> seeds 0..15, 1 -> 16..19
    const int b    = blockIdx.y;
    const int tid  = threadIdx.x;
    const int lane = tid & 31;
    const int half = lane & 15;
    const int wave = tid >> 5;
    const bool lo  = lane < 16;

    const float* p = pcs + (size_t)b * NPTS * 3;

    // ---- B matrix (seeds), loop-invariant; branchless ------------------------
    int sg = g * 16 + half;
    float sx, sy, sz;
    if (sg < NSEEDS) {                       // uniform-ish per block half; cheap
        const float* sp = seeds + ((size_t)b * NSEEDS + sg) * 3;
        sx = sp[0]; sy = sp[1]; sz = sp[2];
    } else {                                 // padding seeds: far away, finite
        sx = 1.0e5f; sy = 1.0e5f; sz = 1.0e5f;
    }
    const float sn = sx * sx + sy * sy + sz * sz;   // |s_n|^2, n = lane&15
    v2f bmat;
    bmat[0] = lo ? sx : sz;
    bmat[1] = lo ? sy : 1.0f;

    float t11[11];
#pragma unroll
    for (int k = 0; k < 11; ++k) t11[k] = 3.0e38f;

    const int ntiles = NPTS >> 4;        // 4096

    for (int T = wave; T < ntiles; T += 8) {
        int pidx = (T << 4) + half;
        const float* pp = p + (size_t)pidx * 3;
        __builtin_prefetch(pp + 16 * 8 * 3, 0, 0);   // this wave's next tile

        // all lanes load the full point (lanes m / m+16 share cache lines)
        float x = pp[0], y = pp[1], z = pp[2];
        float pn = x * x + y * y + z * z;            // |p|^2

        v2f amat;
        amat[0] = lo ? (-2.0f * x) : (-2.0f * z);
        amat[1] = lo ? (-2.0f * y) : pn;

        v8f c = {};
        v8f dotv = __builtin_amdgcn_wmma_f32_16x16x4_f32(
            false, amat, false, bmat, (short)0, c, false, false);

#pragma unroll
        for (int v = 0; v < 8; ++v) {
            float d2 = dotv[v] + sn;                 // squared distance (may be ~-eps at self)
            if (d2 < t11[10]) insert11(t11, d2);
        }
    }

    // ---- merge per-lane top-11 (of d^2) across the block ---------------------
    __shared__ float buf[256 * 11];
#pragma unroll
    for (int k = 0; k < 11; ++k) buf[tid * 11 + k] = t11[k];
    __syncthreads();

    if (tid < 16) {
        float arr[11];
#pragma unroll
        for (int k = 0; k < 11; ++k) arr[k] = 3.0e38f;
        // 16 contributors per seed: threads {w*32 + h*16 + tid}
        for (int w = 0; w < 8; ++w) {
            for (int h = 0; h < 2; ++h) {
                int src = w * 32 + h * 16 + tid;
                for (int k = 0; k < 11; ++k) {
                    float v = buf[src * 11 + k];
                    if (v >= arr[10]) break;   // source rows are sorted ascending
                    insert11(arr, v);
                }
            }
        }
        int seed_id = g * 16 + tid;
        if (seed_id < NSEEDS) {
            // arr[0] is the self distance (~0): dropped, keep arr[1..10]
            float acc = 0.0f;
            for (int k = 1; k <= KNN; ++k) {
                float d2 = fmaxf(arr[k], 0.0f);
                float d  = sqrtf(d2);
                float w  = expf(-d2 / HBW2);
                acc += -d * w;
            }
            atomicAdd(out, acc * inv_b);
        }
    }
}

// ---------------------------------------------------------------------------
extern "C" void kernel_launch(void* const* d_in, const int* in_sizes, int n_in,
                              void* d_out, int out_size, void* d_ws, size_t ws_size,
                              hipStream_t stream) {
    const float* pcs = (const float*)d_in[0];
    const int B = in_sizes[0] / (NPTS * 3);

    float* seeds = (float*)d_ws;     // [B][NSEEDS][3] floats
    float* out   = (float*)d_out;    // scalar

    fps_kernel<<<B, 1024, 0, stream>>>(pcs, seeds, out);

    dim3 grid2(2, B);
    knn_loss_kernel<<<grid2, 256, 0, stream>>>(pcs, seeds, out, 1.0f / (float)B);
}